// SIGRegLoss_1700807049770
// MI455X (gfx1250) — compile-verified
//
#include <hip/hip_runtime.h>

typedef __attribute__((ext_vector_type(2))) float v2f;
typedef __attribute__((ext_vector_type(8))) float v8f;

#define B_DIM 4096
#define D_DIM 2048
#define S_DIM 2048
#define NK 19  // knots 1..19 (knot 0 contributes exactly zero error)

// invnorm[s] = 1 / ||A[:,s]||
__global__ __launch_bounds__(256) void colnorm_kernel(const float* __restrict__ Amat,
                                                      float* __restrict__ invnorm) {
    int s = blockIdx.x * blockDim.x + threadIdx.x;
    if (s >= S_DIM) return;
    float acc = 0.f;
    for (int d = 0; d < D_DIM; ++d) {
        float v = Amat[(size_t)d * S_DIM + s];
        acc = fmaf(v, v, acc);
    }
    invnorm[s] = rsqrtf(acc);
}

// Each wave: 16x32 proj tile (two 16x16 WMMA accumulators sharing the A-frag)
// via V_WMMA_F32_16X16X4_F32 over K=D, then native-cos/sin Chebyshev recurrence
// and LDS/global atomic reduction of the ECF sums.
__global__ __launch_bounds__(256) void gemm_cf_kernel(const float* __restrict__ E,
                                                      const float* __restrict__ Amat,
                                                      const float* __restrict__ t,
                                                      const float* __restrict__ invnorm,
                                                      float* __restrict__ ecf_acc) {
    __shared__ float red[32][NK][2];

    const int tid   = threadIdx.x;
    const int wave  = tid >> 5;
    const int lane  = tid & 31;
    const int lhalf = lane >> 4;   // selects K pair within the 16x16x4 frags
    const int lmod  = lane & 15;

    // zero LDS reduction buffer
    for (int i = tid; i < 32 * NK * 2; i += 256) ((float*)red)[i] = 0.f;
    __syncthreads();

    const int colBase = blockIdx.x * 32;               // 32 slice columns per block
    const int rowBase = blockIdx.y * 128 + wave * 16;  // 16 batch rows per wave
    const int rowA    = rowBase + lmod;                // A-frag: M = lane%16
    const int col0    = colBase + lmod;                // B-frag 0: N = lane%16
    const int col1    = col0 + 16;                     // B-frag 1

    v8f acc0 = {0.f, 0.f, 0.f, 0.f, 0.f, 0.f, 0.f, 0.f};
    v8f acc1 = {0.f, 0.f, 0.f, 0.f, 0.f, 0.f, 0.f, 0.f};
    const float* erow = E + (size_t)rowA * D_DIM;

    #pragma unroll 4
    for (int d = 0; d < D_DIM; d += 4) {
        const int kb = d + lhalf * 2;  // lanes 0-15 -> K={0,1}, lanes 16-31 -> K={2,3}
        v2f a = *(const v2f*)(erow + kb);
        const float* brow0 = Amat + (size_t)kb * S_DIM;
        const float* brow1 = brow0 + S_DIM;
        v2f b0, b1;
        b0.x = brow0[col0];
        b0.y = brow1[col0];
        b1.x = brow0[col1];
        b1.y = brow1[col1];
        acc0 = __builtin_amdgcn_wmma_f32_16x16x4_f32(
            false, a, false, b0, (short)0, acc0, false, false);
        acc1 = __builtin_amdgcn_wmma_f32_16x16x4_f32(
            false, a, false, b1, (short)0, acc1, false, false);
    }

    const float dt = t[1];  // linspace step

    #pragma unroll
    for (int h = 0; h < 2; ++h) {
        const int c = colBase + h * 16 + lmod;
        const float inv = invnorm[c];

        float pv[8];
        #pragma unroll
        for (int v = 0; v < 8; ++v) pv[v] = (h == 0) ? acc0[v] : acc1[v];

        float lc[NK], ls[NK];
        #pragma unroll
        for (int k = 0; k < NK; ++k) { lc[k] = 0.f; ls[k] = 0.f; }

        #pragma unroll
        for (int v = 0; v < 8; ++v) {
            const float th = pv[v] * inv * dt;   // small angle: native sin/cos is fine
            const float c1 = __cosf(th);
            const float s1 = __sinf(th);
            const float tc = 2.f * c1;
            float cp = 1.f, sp = 0.f;  // k=0
            float cc = c1, ss = s1;    // k=1
            lc[0] += cc; ls[0] += ss;
            #pragma unroll
            for (int k = 1; k < NK; ++k) {
                float cn = fmaf(tc, cc, -cp);
                float sn = fmaf(tc, ss, -sp);
                cp = cc; sp = ss; cc = cn; ss = sn;
                lc[k] += cc; ls[k] += ss;
            }
        }

        // cross-wave reduction per output column via LDS float atomics (ds_add_f32)
        #pragma unroll
        for (int k = 0; k < NK; ++k) {
            atomicAdd(&red[h * 16 + lmod][k][0], lc[k]);
            atomicAdd(&red[h * 16 + lmod][k][1], ls[k]);
        }
    }
    __syncthreads();

    // one global atomic per (col, knot, re/im) per block
    for (int i = tid; i < 32 * NK * 2; i += 256) {
        const int n = i / (NK * 2);
        const int r = i % (NK * 2);
        atomicAdd(&ecf_acc[(size_t)(colBase + n) * (NK * 2) + r], ((float*)red)[i]);
    }
}

__global__ __launch_bounds__(256) void finalize_kernel(const float* __restrict__ ecf_acc,
                                                       const float* __restrict__ phi,
                                                       const float* __restrict__ w,
                                                       float* __restrict__ out) {
    __shared__ float red[256];
    const int tid = threadIdx.x;
    const float invB = 1.0f / (float)B_DIM;
    float total = 0.f;
    for (int idx = tid; idx < S_DIM * NK; idx += 256) {
        const int knot = (idx % NK) + 1;
        const float cr = ecf_acc[idx * 2 + 0] * invB;
        const float ci = ecf_acc[idx * 2 + 1] * invB;
        const float dr = cr - phi[knot];
        const float err = dr * dr + ci * ci;
        total = fmaf(err, w[knot], total);
    }
    red[tid] = total;
    __syncthreads();
    for (int off = 128; off > 0; off >>= 1) {
        if (tid < off) red[tid] += red[tid + off];
        __syncthreads();
    }
    if (tid == 0) out[0] = red[0] * ((float)D_DIM / (float)S_DIM);
}

extern "C" void kernel_launch(void* const* d_in, const int* in_sizes, int n_in,
                              void* d_out, int out_size, void* d_ws, size_t ws_size,
                              hipStream_t stream) {
    (void)in_sizes; (void)n_in; (void)out_size; (void)ws_size;
    const float* E    = (const float*)d_in[0];  // [B, D]
    const float* Amat = (const float*)d_in[1];  // [D, S]
    const float* t    = (const float*)d_in[2];  // [20]
    const float* phi  = (const float*)d_in[3];  // [20]
    const float* w    = (const float*)d_in[4];  // [20]
    float* out = (float*)d_out;

    float* invnorm = (float*)d_ws;       // S floats
    float* ecf     = invnorm + S_DIM;    // S * NK * 2 floats

    hipMemsetAsync(d_ws, 0, (size_t)(S_DIM + S_DIM * NK * 2) * sizeof(float), stream);

    colnorm_kernel<<<S_DIM / 256, 256, 0, stream>>>(Amat, invnorm);
    gemm_cf_kernel<<<dim3(S_DIM / 32, B_DIM / 128), 256, 0, stream>>>(E, Amat, t, invnorm, ecf);
    finalize_kernel<<<1, 256, 0, stream>>>(ecf, phi, w, out);
}